// LIF_27513560498762
// MI455X (gfx1250) — compile-verified
//
#include <hip/hip_runtime.h>

// LIF scan over T for x:[B,T,N] f32 -> spikes:[B,T,N] f32.
// HBM-streaming kernel: 537 MB traffic, ~23 us roofline at 23.3 TB/s.
// WMMA inapplicable (nonlinear recurrence: threshold + reset-to-zero).
// This revision adds an explicit register double-buffer software pipeline so
// the next window's 8 global_load_b128 (TH_NT) are in flight while the
// current window computes (partial s_wait_loadcnt instead of wait-0).

typedef float fx4 __attribute__((ext_vector_type(4)));

#define LIF_B 64
#define LIF_T 128
#define LIF_N 8192
#define LIF_VPR (LIF_N / 4)        // 2048 float4 groups per (b,t) row
#define LIF_U 8                    // pipeline window (t-steps per stage)

__device__ __forceinline__ fx4 lif_step(fx4& mem, fx4 cur) {
    fx4 spk;
    #pragma unroll
    for (int i = 0; i < 4; ++i) {
        float m  = 0.25f * mem[i] + cur[i];   // leak-integrate
        bool fire = m > 1.0f;                 // == (m - 1.0f) > 0 exactly (Sterbenz)
        spk[i]   = fire ? 1.0f : 0.0f;
        mem[i]   = fire ? 0.0f : m;           // reset-to-zero on spike
    }
    return spk;
}

__global__ __launch_bounds__(256) void LIF_scan_kernel(
    const float* __restrict__ x, float* __restrict__ out)
{
    // Exact launch: gridDim.x * 256 == B * N/4, no guard needed.
    const unsigned tid = blockIdx.x * 256u + threadIdx.x;
    const unsigned b   = tid >> 11;            // tid / 2048
    const unsigned v   = tid & (LIF_VPR - 1);  // tid % 2048

    const fx4* __restrict__ xp =
        (const fx4*)(x   + (size_t)b * (LIF_T * LIF_N)) + v;
    fx4* __restrict__ op =
        (fx4*)      (out + (size_t)b * (LIF_T * LIF_N)) + v;

    fx4 mem = (fx4)0.0f;

    // Prologue: fill first window.
    fx4 buf[LIF_U];
    #pragma unroll
    for (int i = 0; i < LIF_U; ++i)
        buf[i] = __builtin_nontemporal_load(xp + (unsigned)i * LIF_VPR);

    // Steady state: issue window k+1 loads BEFORE consuming window k.
    for (int t0 = 0; t0 < LIF_T - LIF_U; t0 += LIF_U) {
        fx4 nxt[LIF_U];
        #pragma unroll
        for (int i = 0; i < LIF_U; ++i)
            nxt[i] = __builtin_nontemporal_load(
                xp + (unsigned)(t0 + LIF_U + i) * LIF_VPR);

        #pragma unroll
        for (int i = 0; i < LIF_U; ++i) {
            // gfx1250 prefetch (global_prefetch_b8), 2 windows ahead.
            if (t0 + i + 2 * LIF_U < LIF_T)
                __builtin_prefetch(
                    (const void*)(xp + (unsigned)(t0 + i + 2 * LIF_U) * LIF_VPR), 0, 1);
            fx4 spk = lif_step(mem, buf[i]);
            __builtin_nontemporal_store(spk, op + (unsigned)(t0 + i) * LIF_VPR);
        }

        #pragma unroll
        for (int i = 0; i < LIF_U; ++i) buf[i] = nxt[i];
    }

    // Epilogue: last window (no further loads).
    #pragma unroll
    for (int i = 0; i < LIF_U; ++i) {
        fx4 spk = lif_step(mem, buf[i]);
        __builtin_nontemporal_store(
            spk, op + (unsigned)(LIF_T - LIF_U + i) * LIF_VPR);
    }
}

extern "C" void kernel_launch(void* const* d_in, const int* in_sizes, int n_in,
                              void* d_out, int out_size, void* d_ws, size_t ws_size,
                              hipStream_t stream) {
    const float* x   = (const float*)d_in[0];
    float*       out = (float*)d_out;
    (void)in_sizes; (void)n_in; (void)out_size; (void)d_ws; (void)ws_size;

    // One thread per float4 column: 64 * 2048 = 131072 threads
    // = 512 blocks of 256 (8 wave32s each) -> 4096 waves.
    const unsigned total = LIF_B * LIF_VPR;
    const unsigned block = 256;
    const unsigned nblk  = total / block;    // exact: 512

    LIF_scan_kernel<<<dim3(nblk), dim3(block), 0, stream>>>(x, out);
}